// LastAggregator_91147795956273
// MI455X (gfx1250) — compile-verified
//
#include <hip/hip_runtime.h>

#define WAVES_PER_BLOCK 8
#define WAVE 32

// ---------------------------------------------------------------------------
// Pass 0: zero the per-node packed keys in workspace (must re-init every call;
// harness does not re-poison between replays).
// ---------------------------------------------------------------------------
__global__ void la_zero_keys(unsigned long long* __restrict__ keys, int n) {
  int i = blockIdx.x * blockDim.x + threadIdx.x;
  if (i < n) keys[i] = 0ull;
}

// ---------------------------------------------------------------------------
// Pass 1: per-event scatter.  key = (t << 32) | (event_id + 1).
// atomicMax on u64 gives: max t first, then max event id (reference
// tie-break), and key==0 <=> node received no events.
// ---------------------------------------------------------------------------
__global__ void la_scatter_max(const int* __restrict__ index,
                               const int* __restrict__ t,
                               unsigned long long* __restrict__ keys,
                               int E) {
  int e = blockIdx.x * blockDim.x + threadIdx.x;
  if (e < E) {
    unsigned long long key =
        ((unsigned long long)(unsigned)t[e] << 32) | (unsigned)(e + 1);
    atomicMax(&keys[index[e]], key);  // -> global_atomic_max_u64 (L2 atomic)
  }
}

// ---------------------------------------------------------------------------
// Pass 2: one wave32 per output row (D=128 floats = 32 lanes x B128).
// Valid rows are copied with the CDNA5 async global<->LDS DMA path:
//   global_load_async_to_lds_b128 -> s_wait_asynccnt 0 ->
//   global_store_async_from_lds_b128
// so row data never occupies VGPRs; the split ASYNCcnt counter sequences the
// LDS handoff.  Latency is hidden by 100k independent waves.
// Invalid rows store zeros directly.
// ---------------------------------------------------------------------------
__global__ __launch_bounds__(WAVES_PER_BLOCK * WAVE) void la_gather(
    const float* __restrict__ msg,
    const unsigned long long* __restrict__ keys,
    float* __restrict__ out, int N, int D) {
  // 512 B staging buffer per wave (32 lanes * 16 B)
  __shared__ float lds_buf[WAVES_PER_BLOCK * WAVE * 4];

  const int lane = threadIdx.x & (WAVE - 1);
  const int wib  = threadIdx.x / WAVE;               // wave index in block
  const int row  = blockIdx.x * WAVES_PER_BLOCK + wib;
  if (row >= N) return;                              // wave-uniform exit

  const unsigned long long key = keys[row];
  float* dst_row = out + (size_t)row * D;

  if (key == 0ull) {
    // Node has no events: write zeros (poisoned d_out must be overwritten).
    for (int c = lane * 4; c < D; c += WAVE * 4) {
      float4 z = make_float4(0.f, 0.f, 0.f, 0.f);
      *(float4*)(dst_row + c) = z;
    }
    return;
  }

  const unsigned src_row_i = (unsigned)(key & 0xffffffffu) - 1u;
  const float* src_row = msg + (size_t)src_row_i * D;

  // Low 32 bits of the generic shared-aperture address == LDS byte offset.
  unsigned lds_off =
      (unsigned)(unsigned long long)(&lds_buf[wib * WAVE * 4]) +
      (unsigned)(lane * 16);

  for (int c = lane * 4; c < D; c += WAVE * 4) {      // one iter for D=128
    const float* gsrc = src_row + c;
    float*       gdst = dst_row + c;
    asm volatile("global_load_async_to_lds_b128 %0, %1, off"
                 :: "v"(lds_off), "v"(gsrc)
                 : "memory");
    asm volatile("s_wait_asynccnt 0" ::: "memory");    // LDS now holds the row
    asm volatile("global_store_async_from_lds_b128 %0, %1, off"
                 :: "v"(gdst), "v"(lds_off)
                 : "memory");
    asm volatile("s_wait_asynccnt 0" ::: "memory");    // LDS reusable
  }
}

// ---------------------------------------------------------------------------
// Host launcher.  Inputs: msg [E,D] f32, index [E] i32, t [E] i32, dim_size.
// Sizes derived from in_sizes/out_size only (graph-capture safe, no d2h).
// ---------------------------------------------------------------------------
extern "C" void kernel_launch(void* const* d_in, const int* in_sizes, int n_in,
                              void* d_out, int out_size, void* d_ws,
                              size_t ws_size, hipStream_t stream) {
  const float* msg   = (const float*)d_in[0];
  const int*   index = (const int*)d_in[1];
  const int*   t     = (const int*)d_in[2];

  const int E = in_sizes[1];                 // number of events
  const int D = in_sizes[0] / in_sizes[1];   // feature dim (128)
  const int N = out_size / D;                // number of nodes

  unsigned long long* keys = (unsigned long long*)d_ws;  // N * 8 bytes
  float* out = (float*)d_out;

  {
    int threads = 256;
    int blocks  = (N + threads - 1) / threads;
    la_zero_keys<<<blocks, threads, 0, stream>>>(keys, N);
  }
  {
    int threads = 256;
    int blocks  = (E + threads - 1) / threads;
    la_scatter_max<<<blocks, threads, 0, stream>>>(index, t, keys, E);
  }
  {
    int threads = WAVES_PER_BLOCK * WAVE;    // 256 = 8 waves
    int blocks  = (N + WAVES_PER_BLOCK - 1) / WAVES_PER_BLOCK;
    la_gather<<<blocks, threads, 0, stream>>>(msg, keys, out, N, D);
  }
}